// DiffeomorphicTransformer_18176301597380
// MI455X (gfx1250) — compile-verified
//
#include <hip/hip_runtime.h>

// ---------------------------------------------------------------------------
// CPAB diffeomorphic transformer, MI455X (gfx1250, wave32)
//   Stage 0: zero-pad params -> [32][28], basis -> [224][28]   (prep, tiny)
//   Stage 1: Avees = params @ basis^T via V_WMMA_F32_16X16X4_F32, branch-free
//   Stage 2: fused 50-step Euler integration (affine table in LDS) + bilinear
// ---------------------------------------------------------------------------

#define NXc     3
#define NYc     3
#define NTt     36              // triangles
#define NSTEPS  50
#define NB      32              // batch
#define NC      3               // channels
#define HH      256
#define WW      256
#define NPIX    (HH * WW)
#define Dd      26              // theta dim
#define NCOLS   (NTt * 6)       // 216 = NT*2*3
#define KP      28              // K padded to multiple of 4
#define NP      224             // N padded to multiple of 16

typedef __attribute__((ext_vector_type(2)))  float v2f;
typedef __attribute__((ext_vector_type(8)))  float v8f;

// Workspace layout (floats):
//   [0,      896)   params_pad [32][28]
//   [896,   7168)   basis_pad  [224][28]
//   [7168, 14336)   avees_pad  [32][224]
#define WS_PARAMS 0
#define WS_BASIS  (NB * KP)                 // 896
#define WS_AVEES  (WS_BASIS + NP * KP)      // 7168

// ---------------------------------------------------------------------------
// Stage 0: zero-padded staging copies so the WMMA kernel is branch-free.
// ---------------------------------------------------------------------------
__global__ __launch_bounds__(256) void pad_inputs(
    const float* __restrict__ params,   // [32][26]
    const float* __restrict__ basis,    // [216][26]
    float* __restrict__ ws)
{
    const int t = blockIdx.x * 256 + threadIdx.x;
    const int nA = NB * KP;             // 896
    const int nB = NP * KP;             // 6272
    if (t < nA) {
        const int m = t / KP, k = t % KP;
        ws[WS_PARAMS + t] = (k < Dd) ? params[m * Dd + k] : 0.0f;
    } else if (t < nA + nB) {
        const int u = t - nA;
        const int n = u / KP, k = u % KP;
        ws[WS_BASIS + u] = (n < NCOLS && k < Dd) ? basis[n * Dd + k] : 0.0f;
    }
}

// ---------------------------------------------------------------------------
// Stage 1: C[32 x 224] = params_pad[32 x 28] @ basis_pad[224 x 28]^T
// One wave32 per 16x16 output tile: gridDim.x = 2 * 14 = 28 blocks of 32 thr.
// EXEC is all-ones for the whole kernel (WMMA requirement); no predication.
// Fragment loads are single aligned global_load_b64 each.
// ---------------------------------------------------------------------------
__global__ __launch_bounds__(32) void avees_gemm_wmma(
    const float* __restrict__ ws)       // padded staging + output
{
    const float* params_pad = ws + WS_PARAMS;   // [32][28]
    const float* basis_pad  = ws + WS_BASIS;    // [224][28]
    float*       avees_pad  = const_cast<float*>(ws) + WS_AVEES;  // [32][224]

    const int lane  = threadIdx.x;      // 0..31
    const int l16   = lane & 15;
    const int half  = lane >> 4;        // 0 | 1
    const int mtile = blockIdx.x / 14;  // 2 M-tiles (M=32)
    const int ntile = blockIdx.x % 14;  // 14 N-tiles (N=224)

    const int m = mtile * 16 + l16;     // A row   (lanes 0-15 / 16-31 both M=l16)
    const int n = ntile * 16 + l16;     // B column

    v8f acc = {};

    // A 16x4 f32 layout: lane l -> M = l%16, K = 2*(l/16) + vgpr
    // B  4x16 f32 layout (row-striped like C/D): lane l -> N = l%16, K = 2*(l/16) + vgpr
    #pragma unroll
    for (int k0 = 0; k0 < KP; k0 += 4) {
        const int ka = k0 + half * 2;   // even -> 8B-aligned v2f loads
        const v2f a  = *reinterpret_cast<const v2f*>(params_pad + m * KP + ka);
        const v2f bb = *reinterpret_cast<const v2f*>(basis_pad  + n * KP + ka);
        acc = __builtin_amdgcn_wmma_f32_16x16x4_f32(
                  false, a, false, bb, (short)0, acc, false, false);
    }

    // C/D 16x16 f32 layout: VGPR v, lanes 0-15 -> M=v, lanes 16-31 -> M=v+8
    #pragma unroll
    for (int v = 0; v < 8; ++v) {
        const int mr = mtile * 16 + half * 8 + v;
        avees_pad[mr * NP + n] = acc[v];
    }
}

// ---------------------------------------------------------------------------
// Stage 2: fused CPAB integration + bilinear sampling.
// One thread per (batch, pixel); 256 consecutive pixels of one batch per
// block -> coalesced output stores. Per-batch 216-float affine table in LDS,
// gathered with ds_load_b96 pairs each Euler step (zero HBM traffic in the
// hot loop). Image gathers hit L2 (whole input set = 25 MB << 192 MB L2).
// ---------------------------------------------------------------------------
__global__ __launch_bounds__(256) void cpab_warp_sample(
    const float* __restrict__ img,      // [32][3][256][256]
    const float* __restrict__ ws,       // avees_pad at WS_AVEES, [32][224]
    float* __restrict__ out)            // [32][3][256][256]
{
    __shared__ float sA[NCOLS];
    const float* avees_pad = ws + WS_AVEES;

    const int bpb = NPIX / 256;                         // 256 blocks per batch
    const int b   = blockIdx.x / bpb;
    const int pix = (blockIdx.x % bpb) * 256 + threadIdx.x;

    if (threadIdx.x < NCOLS) sA[threadIdx.x] = avees_pad[b * NP + threadIdx.x];
    __syncthreads();

    const int h = pix >> 8;
    const int w = pix & 255;
    float px = (float)w * (1.0f / 255.0f);              // grid x in [0,1]
    float py = (float)h * (1.0f / 255.0f);              // grid y in [0,1]

    const float eps  = 1e-6f;
    const float invN = 1.0f / (float)NSTEPS;

    #pragma unroll 2
    for (int s = 0; s < NSTEPS; ++s) {
        const float xc = fminf(fmaxf(px, 0.0f), 1.0f - eps);
        const float yc = fminf(fmaxf(py, 0.0f), 1.0f - eps);
        const float fx = xc * (float)NXc;
        const float fy = yc * (float)NYc;
        const float fi = floorf(fx);
        const float fj = floorf(fy);
        const float u  = fx - fi;
        const float wv = fy - fj;
        // nested jnp.where: cond(tri=0) outranks cond(tri=2) outranks u-test
        int tri = (u >= 0.5f) ? 1 : 3;
        tri = ((wv >= u) && (wv >= 1.0f - u)) ? 2 : tri;
        tri = ((wv <= u) && (wv <= 1.0f - u)) ? 0 : tri;
        const int idx = ((((int)fj) * NXc + (int)fi) * 4 + tri) * 6;
        const float vx = sA[idx + 0] * px + sA[idx + 1] * py + sA[idx + 2];
        const float vy = sA[idx + 3] * px + sA[idx + 4] * py + sA[idx + 5];
        px += vx * invN;
        py += vy * invN;
    }

    // Bilinear sample: x0 clamped to [0, W-2] so x0+1 stays in range.
    const float gx = fminf(fmaxf(px, 0.0f), 1.0f) * (float)(WW - 1);
    const float gy = fminf(fmaxf(py, 0.0f), 1.0f) * (float)(HH - 1);
    int x0 = (int)floorf(gx); x0 = min(max(x0, 0), WW - 2);
    int y0 = (int)floorf(gy); y0 = min(max(y0, 0), HH - 2);
    const float frx = gx - (float)x0;
    const float fry = gy - (float)y0;
    const float w00 = (1.0f - frx) * (1.0f - fry);
    const float w01 = frx * (1.0f - fry);
    const float w10 = (1.0f - frx) * fry;
    const float w11 = frx * fry;
    const int base = y0 * WW + x0;

    #pragma unroll
    for (int c = 0; c < NC; ++c) {
        const float* p = img + (b * NC + c) * NPIX;
        const float v00 = p[base];
        const float v01 = p[base + 1];
        const float v10 = p[base + WW];
        const float v11 = p[base + WW + 1];
        out[(b * NC + c) * NPIX + pix] =
            w00 * v00 + w01 * v01 + w10 * v10 + w11 * v11;
    }
}

// ---------------------------------------------------------------------------
extern "C" void kernel_launch(void* const* d_in, const int* in_sizes, int n_in,
                              void* d_out, int out_size, void* d_ws, size_t ws_size,
                              hipStream_t stream) {
    (void)in_sizes; (void)n_in; (void)out_size; (void)ws_size;
    const float* x      = (const float*)d_in[0];   // [32][3][256][256]
    const float* params = (const float*)d_in[1];   // [32][26]
    const float* basis  = (const float*)d_in[2];   // [216][26]
    float*       out    = (float*)d_out;           // [32][3][256][256]
    float*       ws     = (float*)d_ws;            // 14336 floats (56 KB)

    pad_inputs<<<28, 256, 0, stream>>>(params, basis, ws);        // 7168 elems
    avees_gemm_wmma<<<28, 32, 0, stream>>>(ws);                    // 28 tiles
    cpab_warp_sample<<<NB * (NPIX / 256), 256, 0, stream>>>(x, ws, out);
}